// FiniteSeparableModel_71897752535165
// MI455X (gfx1250) — compile-verified
//
#include <hip/hip_runtime.h>
#include <hip/hip_bf16.h>
#include <stdint.h>

// Problem constants (match reference)
#define NXX 4001
#define NYY 2001
#define DD  8

// -------- TDM descriptor helpers (CDNA5 Tensor Data Mover) --------
typedef unsigned int u32x4 __attribute__((ext_vector_type(4)));
typedef int          i32x8 __attribute__((ext_vector_type(8)));
typedef int          i32x4 __attribute__((ext_vector_type(4)));

// Low 32 bits of a flat pointer into LDS aperture == LDS byte offset
// (ISA 10.2: LDS aperture match uses addr[63:32]; addr[31:0] is the offset).
__device__ __forceinline__ unsigned lds_byte_off(const void* p) {
  return (unsigned)(uintptr_t)p;
}

// 2D (or 1D when tile_d1==1) tile load: global -> LDS via TDM.
// Elements are 4-byte. tensor_d0/d1 are the REMAINING extents measured from
// the tile start, so out-of-tensor reads are suppressed (zero-filled) and we
// never touch global memory past the buffer.  pad!=0 inserts 1 dword of LDS
// padding after every 256 dwords (row stride 257) for bank-conflict-free reads.
__device__ __forceinline__ void tdm_load_f32_tile(
    unsigned lds_off, const void* gptr,
    unsigned tensor_d0, unsigned tensor_d1,
    unsigned tile_d0, unsigned tile_d1,
    unsigned stride0_elems, int pad)
{
  uint64_t ga = (uint64_t)(uintptr_t)gptr;
  u32x4 g0;
  g0.x = 1u;                                            // count=1, user mode
  g0.y = lds_off;                                       // lds_addr (bytes)
  g0.z = (unsigned)(ga & 0xFFFFFFFFu);                  // global_addr[31:0]
  g0.w = (unsigned)((ga >> 32) & 0x1FFFFFFu)            // global_addr[56:32]
       | (2u << 30);                                    // type = 2 ("image")
  unsigned w0 = (2u << 16);                             // data_size = 4 bytes
  if (pad) w0 |= (1u << 20) | (7u << 22);               // pad_en, every 256dw, +1dw
  i32x8 g1;
  g1[0] = (int)w0;                                      // wg_mask=0 (no cluster)
  g1[1] = (int)((tensor_d0 & 0xFFFFu) << 16);           // tensor_dim0[15:0]
  g1[2] = (int)(((tensor_d0 >> 16) & 0xFFFFu) | ((tensor_d1 & 0xFFFFu) << 16));
  g1[3] = (int)(((tensor_d1 >> 16) & 0xFFFFu) | ((tile_d0 & 0xFFFFu) << 16));
  g1[4] = (int)(tile_d1 & 0xFFFFu);                     // tile_dim1, tile_dim2=0
  g1[5] = (int)stride0_elems;                           // tensor_dim0_stride[31:0]
  g1[6] = 0;
  g1[7] = 0;
  i32x4 gz4 = {0, 0, 0, 0};                             // 2D tensor: groups 2/3 unused
  i32x8 gz8 = {0, 0, 0, 0, 0, 0, 0, 0};                 // extra operand (unused here)
  // clang-23 / therock toolchain: 6-argument form
  __builtin_amdgcn_tensor_load_to_lds(g0, g1, gz4, gz4, gz8, 0);
}

// -------- order-preserving f32 <-> u32 encoding (exact atomic max) --------
__device__ __forceinline__ unsigned enc_f32(float f) {
  unsigned u = __float_as_uint(f);
  return (u & 0x80000000u) ? ~u : (u | 0x80000000u);
}
__device__ __forceinline__ float dec_f32(unsigned e) {
  unsigned u = (e & 0x80000000u) ? (e ^ 0x80000000u) : ~e;
  return __uint_as_float(u);
}

// =======================================================================
// Phase 1: u[x][d] = max_y (K[x,y] - b[y,d]).  One WG per 32 x-rows.
// =======================================================================
#define XT   32
#define YT   256
#define KROW (YT + 1)   // 257-dword padded row stride (TDM pad)

__global__ __launch_bounds__(256)
void lft_phase1_u(const float* __restrict__ K, const float* __restrict__ b,
                  float* __restrict__ u)
{
  __shared__ __align__(16) float sK[2][XT * KROW];   // 2 x 32.1 KB
  __shared__ __align__(16) float sB[2][YT * DD];     // 2 x 8 KB

  const int tid = threadIdx.x;
  const int x0  = blockIdx.x * XT;
  const int r   = tid & 31;    // row within tile (lane)
  const int s   = tid >> 5;    // y-slice (wave id), 8 slices x 32 y

  float m[DD];
#pragma unroll
  for (int d = 0; d < DD; ++d) m[d] = -__builtin_inff();

  const int nchunk = (NYY + YT - 1) / YT;  // 8

  if (tid < 32) {  // wave 0 drives the TDM (EXEC-ignored wave instruction)
    tdm_load_f32_tile(lds_byte_off(&sK[0][0]), K + (size_t)x0 * NYY,
                      (unsigned)NYY, (unsigned)(NXX - x0), YT, XT, NYY, 1);
    tdm_load_f32_tile(lds_byte_off(&sB[0][0]), b,
                      (unsigned)NYY * DD, 1u, YT * DD, 1u, YT * DD, 0);
  }

  for (int c = 0; c < nchunk; ++c) {
    const int buf = c & 1;
    const int y0  = c * YT;
    if (tid < 32) {
      if (c + 1 < nchunk) {
        const int y1 = (c + 1) * YT;
        tdm_load_f32_tile(lds_byte_off(&sK[buf ^ 1][0]),
                          K + (size_t)x0 * NYY + y1,
                          (unsigned)(NYY - y1), (unsigned)(NXX - x0),
                          YT, XT, NYY, 1);
        tdm_load_f32_tile(lds_byte_off(&sB[buf ^ 1][0]), b + (size_t)y1 * DD,
                          (unsigned)(NYY - y1) * DD, 1u, YT * DD, 1u, YT * DD, 0);
        __builtin_amdgcn_s_wait_tensorcnt((short)2);   // chunk c resident
      } else {
        __builtin_amdgcn_s_wait_tensorcnt((short)0);
      }
    }
    __syncthreads();

    const float* kt = &sK[buf][r * KROW];
    const float* bt = &sB[buf][0];
#pragma unroll 4
    for (int i = 0; i < 32; ++i) {
      const int yl = s * 32 + i;
      const int yg = y0 + yl;                 // uniform per wave -> cheap branch
      if (yg < NYY) {
        const float  kv = kt[yl];
        const float4 b0 = *(const float4*)(bt + (size_t)yl * DD);
        const float4 b1 = *(const float4*)(bt + (size_t)yl * DD + 4);
        m[0] = fmaxf(m[0], kv - b0.x);
        m[1] = fmaxf(m[1], kv - b0.y);
        m[2] = fmaxf(m[2], kv - b0.z);
        m[3] = fmaxf(m[3], kv - b0.w);
        m[4] = fmaxf(m[4], kv - b1.x);
        m[5] = fmaxf(m[5], kv - b1.y);
        m[6] = fmaxf(m[6], kv - b1.z);
        m[7] = fmaxf(m[7], kv - b1.w);
      }
    }
    __syncthreads();
  }

  // Reduce the 8 y-slices per (row, dim) via LDS (reuse sB[0]: 2048 floats).
  float* red = &sB[0][0];
#pragma unroll
  for (int d = 0; d < DD; ++d) red[tid * DD + d] = m[d];  // [s][r][d]
  __syncthreads();

  const int r2 = tid >> 3;   // row 0..31
  const int d2 = tid & 7;    // dim 0..7
  float v = red[(0 * 32 + r2) * DD + d2];
#pragma unroll
  for (int ss = 1; ss < 8; ++ss)
    v = fmaxf(v, red[(ss * 32 + r2) * DD + d2]);
  const int xg = x0 + r2;
  if (xg < NXX) u[(size_t)xg * DD + d2] = v;   // coalesced
}

// =======================================================================
// Phase 2: out[y][d] = max_x (K[x,y] - u[x][d]).  WG = 32 y-cols x 4 x-chunks;
// partials combined with exact sortable-uint atomic max.
// =======================================================================
#define YT2 32
#define XC  256

__global__ __launch_bounds__(256)
void lft_phase2_b(const float* __restrict__ K, const float* __restrict__ u,
                  unsigned* __restrict__ outEnc)
{
  __shared__ __align__(16) float sK[2][XC * YT2];    // 2 x 32 KB
  __shared__ __align__(16) float sU[2][XC * DD];     // 2 x 8 KB

  const int tid = threadIdx.x;
  const int y0  = blockIdx.x * YT2;
  const int yl  = tid & 31;    // y within tile (lane)
  const int s   = tid >> 5;    // x-slice (wave id), 8 slices x 32 x
  const int CPG = 4;           // x-chunks per WG (gridDim.y == 4, 16 chunks total)
  const int cbeg = blockIdx.y * CPG;

  float m[DD];
#pragma unroll
  for (int d = 0; d < DD; ++d) m[d] = -__builtin_inff();

  if (tid < 32) {
    const int xb = cbeg * XC;
    tdm_load_f32_tile(lds_byte_off(&sK[0][0]), K + (size_t)xb * NYY + y0,
                      (unsigned)(NYY - y0), (unsigned)(NXX - xb), YT2, XC, NYY, 0);
    tdm_load_f32_tile(lds_byte_off(&sU[0][0]), u + (size_t)xb * DD,
                      (unsigned)(NXX - xb) * DD, 1u, XC * DD, 1u, XC * DD, 0);
  }

  for (int j = 0; j < CPG; ++j) {
    const int buf   = j & 1;
    const int xbase = (cbeg + j) * XC;
    if (tid < 32) {
      if (j + 1 < CPG) {
        const int xb1 = (cbeg + j + 1) * XC;
        tdm_load_f32_tile(lds_byte_off(&sK[buf ^ 1][0]),
                          K + (size_t)xb1 * NYY + y0,
                          (unsigned)(NYY - y0), (unsigned)(NXX - xb1),
                          YT2, XC, NYY, 0);
        tdm_load_f32_tile(lds_byte_off(&sU[buf ^ 1][0]), u + (size_t)xb1 * DD,
                          (unsigned)(NXX - xb1) * DD, 1u, XC * DD, 1u, XC * DD, 0);
        __builtin_amdgcn_s_wait_tensorcnt((short)2);
      } else {
        __builtin_amdgcn_s_wait_tensorcnt((short)0);
      }
    }
    __syncthreads();

    const float* kt = &sK[buf][0];
    const float* ut = &sU[buf][0];
#pragma unroll 4
    for (int i = 0; i < 32; ++i) {
      const int xl = s * 32 + i;
      const int xg = xbase + xl;              // uniform per wave
      if (xg < NXX) {
        const float  kv = kt[xl * YT2 + yl];  // lanes read consecutive dwords
        const float4 u0 = *(const float4*)(ut + (size_t)xl * DD);      // broadcast
        const float4 u1 = *(const float4*)(ut + (size_t)xl * DD + 4);
        m[0] = fmaxf(m[0], kv - u0.x);
        m[1] = fmaxf(m[1], kv - u0.y);
        m[2] = fmaxf(m[2], kv - u0.z);
        m[3] = fmaxf(m[3], kv - u0.w);
        m[4] = fmaxf(m[4], kv - u1.x);
        m[5] = fmaxf(m[5], kv - u1.y);
        m[6] = fmaxf(m[6], kv - u1.z);
        m[7] = fmaxf(m[7], kv - u1.w);
      }
    }
    __syncthreads();
  }

  // Reduce 8 x-slices per (y, dim) via LDS (reuse sU[0]), then atomic-max out.
  float* red = &sU[0][0];
#pragma unroll
  for (int d = 0; d < DD; ++d) red[tid * DD + d] = m[d];
  __syncthreads();

  const int y2 = tid >> 3;
  const int d2 = tid & 7;
  float v = red[(0 * 32 + y2) * DD + d2];
#pragma unroll
  for (int ss = 1; ss < 8; ++ss)
    v = fmaxf(v, red[(ss * 32 + y2) * DD + d2]);
  const int yg = y0 + y2;
  if (yg < NYY) atomicMax(&outEnc[(size_t)yg * DD + d2], enc_f32(v));
}

// -------- tiny init / decode passes over d_out (16008 elems) --------
__global__ void lft_init_out(unsigned* __restrict__ o, int n) {
  int i = blockIdx.x * blockDim.x + threadIdx.x;
  if (i < n) o[i] = 0u;   // 0 < enc(x) for every real x (incl. -inf)
}
__global__ void lft_decode_out(unsigned* __restrict__ o, int n) {
  int i = blockIdx.x * blockDim.x + threadIdx.x;
  if (i < n) o[i] = __float_as_uint(dec_f32(o[i]));
}

extern "C" void kernel_launch(void* const* d_in, const int* in_sizes, int n_in,
                              void* d_out, int out_size, void* d_ws, size_t ws_size,
                              hipStream_t stream) {
  (void)in_sizes; (void)n_in; (void)out_size; (void)ws_size;
  const float* K = (const float*)d_in[0];        // [NX, NY] f32
  const float* b = (const float*)d_in[1];        // [NY, D]  f32
  float*    u    = (float*)d_ws;                 // [NX, D]  f32 scratch (128 KB)
  unsigned* oEnc = (unsigned*)d_out;             // encoded view of d_out
  float*    oF   = (float*)d_out;                // final f32 view (in-place decode)
  (void)oF;

  const int nOut = NYY * DD;                     // 16008
  lft_init_out<<<(nOut + 255) / 256, 256, 0, stream>>>(oEnc, nOut);

  const int g1 = (NXX + XT - 1) / XT;            // 126 WGs
  lft_phase1_u<<<g1, 256, 0, stream>>>(K, b, u);

  dim3 g2((NYY + YT2 - 1) / YT2, 4);             // 63 x 4 = 252 WGs
  lft_phase2_b<<<g2, 256, 0, stream>>>(K, u, oEnc);

  lft_decode_out<<<(nOut + 255) / 256, 256, 0, stream>>>(oEnc, nOut);
}